// QuestAttention_77979426226494
// MI455X (gfx1250) — compile-verified
//
#include <hip/hip_runtime.h>
#include <hip/hip_bf16.h>
#include <math.h>

typedef __attribute__((ext_vector_type(2))) float vf2;
typedef __attribute__((ext_vector_type(8))) float vf8;

constexpr int H      = 32;
constexpr int KV     = 8;
constexpr int Dh     = 128;
constexpr int PAGE   = 16;
constexpr int TOPK   = 256;
constexpr int HID    = H * Dh;          // 4096
constexpr int KVDIM  = KV * Dh;         // 1024
constexpr int SEQ_PAST = 32767;         // tokens in cache
constexpr int NPAGES   = 2048;          // (SEQ_PAST+1)/PAGE
constexpr int NEWTOK   = SEQ_PAST;      // appended token index
constexpr float SM_SCALE  = 0.08838834764831845f;  // 1/sqrt(128)
constexpr float LOG_THETA = 9.210340371976184f;    // ln(10000)

constexpr int GEMV_SPLITS    = 8;
constexpr int ROWS_PER_SPLIT = HID / GEMV_SPLITS;  // 512
constexpr int QKV_STRIDE     = HID + 2 * KVDIM;    // 6144

// ---- workspace layout (float offsets) ----
constexpr size_t OFF_Q      = 0;                                    // 4096 (rotated q)
constexpr size_t OFF_K      = OFF_Q + HID;                          // 1024 (rotated new k)
constexpr size_t OFF_V      = OFF_K + KVDIM;                        // 1024 (new v)
constexpr size_t OFF_SCORES = OFF_V + KVDIM;                        // H*NPAGES
constexpr size_t OFF_SPAGES = OFF_SCORES + (size_t)H * NPAGES;      // H*TOPK ints
constexpr size_t OFF_PART   = OFF_SPAGES + (size_t)H * TOPK;        // H*8*(Dh+2)
constexpr size_t OFF_ATTN   = OFF_PART + (size_t)H * 8 * (Dh + 2);  // 4096
constexpr size_t OFF_GPART  = OFF_ATTN + HID;                       // 8*6144
constexpr size_t OFF_OPART  = OFF_GPART + (size_t)GEMV_SPLITS * QKV_STRIDE; // 8*4096

// =====================================================================
// GEMV with split-K partials (deterministic; no float atomics).
// out[j] = sum_i x[i] * W[i*outDim + j]; W streamed coalesced per row.
// =====================================================================
__global__ __launch_bounds__(256)
void gemv_partial(const float* __restrict__ x, const float* __restrict__ W,
                  float* __restrict__ part, int outDim, int outStride, int colBase)
{
    __shared__ float xs[ROWS_PER_SPLIT];
    const int col = blockIdx.x * 256 + threadIdx.x;
    const int r0  = blockIdx.y * ROWS_PER_SPLIT;
    for (int i = threadIdx.x; i < ROWS_PER_SPLIT; i += 256) xs[i] = x[r0 + i];
    __syncthreads();
    const float* Wp = W + (size_t)r0 * outDim + col;
    float acc = 0.f;
#pragma unroll 8
    for (int i = 0; i < ROWS_PER_SPLIT; ++i) {
        if ((i & 63) == 0) __builtin_prefetch(Wp + (size_t)32 * outDim, 0, 0);
        acc = fmaf(xs[i], *Wp, acc);
        Wp += outDim;
    }
    part[(size_t)blockIdx.y * outStride + colBase + col] = acc;
}

__global__ __launch_bounds__(256)
void reduce_splits(const float* __restrict__ part, float* __restrict__ out,
                   int n, int stride)
{
    int col = blockIdx.x * 256 + threadIdx.x;
    if (col >= n) return;
    float s = 0.f;
#pragma unroll
    for (int k = 0; k < GEMV_SPLITS; ++k) s += part[(size_t)k * stride + col];
    out[col] = s;
}

// =====================================================================
// RoPE in place on q (32 heads) and new k (8 heads) at pos = 32767.
// =====================================================================
__global__ __launch_bounds__(256)
void rope_kernel(float* __restrict__ ws)
{
    int idx = blockIdx.x * 256 + threadIdx.x;
    if (idx >= (H + KV) * (Dh / 2)) return;
    int hh = idx >> 6;           // Dh/2 = 64 pairs per head
    int j  = idx & 63;
    float* base = (hh < H) ? (ws + OFF_Q + (size_t)hh * Dh)
                           : (ws + OFF_K + (size_t)(hh - H) * Dh);
    float inv = __expf(-(float)j * (2.0f / (float)Dh) * LOG_THETA);
    float ang = (float)SEQ_PAST * inv;
    float s, c;
    __sincosf(ang, &s, &c);
    float lo = base[j], hi = base[j + 64];
    base[j]      = lo * c - hi * s;
    base[j + 64] = hi * c + lo * s;
}

// =====================================================================
// Page min/max scan + Quest scores. One block per page; streams the
// page's 64KB fully coalesced (256 lanes x float4 = one token row/iter).
// score[h][p] = sum_d max(q_d*kmax_d, q_d*kmin_d); last page forced +inf.
// =====================================================================
__global__ __launch_bounds__(256)
void page_score_kernel(const float* __restrict__ kc, const float* __restrict__ ws,
                       float* __restrict__ scores)
{
    const int p = blockIdx.x;
    const int tid = threadIdx.x;
    if (p == NPAGES - 1) {                    // reference: score[:, -1] = inf
        if (tid < H) scores[(size_t)tid * NPAGES + p] = INFINITY;
        return;
    }
    const int kvh  = tid >> 5;                // wave w handles kv-head w
    const int lane = tid & 31;
    const int dbase = lane * 4;
    const float* base = kc + (size_t)p * PAGE * KVDIM + kvh * Dh + dbase;
    float4 mx = *(const float4*)base;
    float4 mn = mx;
#pragma unroll
    for (int t = 1; t < PAGE; ++t) {
        float4 v = *(const float4*)(base + (size_t)t * KVDIM);
        mx.x = fmaxf(mx.x, v.x); mn.x = fminf(mn.x, v.x);
        mx.y = fmaxf(mx.y, v.y); mn.y = fminf(mn.y, v.y);
        mx.z = fmaxf(mx.z, v.z); mn.z = fminf(mn.z, v.z);
        mx.w = fmaxf(mx.w, v.w); mn.w = fminf(mn.w, v.w);
    }
    const float* q = ws + OFF_Q;
#pragma unroll
    for (int g = 0; g < 4; ++g) {             // 4 query heads per kv head
        int h = kvh * 4 + g;
        float4 qv = *(const float4*)(q + (size_t)h * Dh + dbase);
        float s = fmaxf(qv.x * mx.x, qv.x * mn.x) + fmaxf(qv.y * mx.y, qv.y * mn.y)
                + fmaxf(qv.z * mx.z, qv.z * mn.z) + fmaxf(qv.w * mx.w, qv.w * mn.w);
#pragma unroll
        for (int off = 16; off > 0; off >>= 1) s += __shfl_xor(s, off, 32);
        if (lane == 0) scores[(size_t)h * NPAGES + p] = s;
    }
}

// =====================================================================
// Exact top-256 of 2048 per head via MSB-first radix select on monotone
// uint keys (order of output is irrelevant: softmax+sum are permutation
// invariant). The +inf page maps to the max key and is always kept.
// =====================================================================
__global__ __launch_bounds__(256)
void topk_kernel(const float* __restrict__ scores, int* __restrict__ spages)
{
    const int h = blockIdx.x;
    const int tid = threadIdx.x;
    __shared__ int cnt;
    __shared__ int wpos;
    unsigned key[8];
    int col[8];
#pragma unroll
    for (int r = 0; r < 8; ++r) {
        int c = r * 256 + tid;
        unsigned u = __float_as_uint(scores[(size_t)h * NPAGES + c]);
        key[r] = (u & 0x80000000u) ? ~u : (u | 0x80000000u);
        col[r] = c;
    }
    unsigned prefix = 0;
    int need = TOPK;
    for (int bit = 31; bit >= 0; --bit) {
        if (tid == 0) cnt = 0;
        __syncthreads();
        unsigned bm = 1u << bit;
        unsigned high = ~(bm | (bm - 1u));    // bits strictly above current
        int loc = 0;
#pragma unroll
        for (int r = 0; r < 8; ++r)
            if ((((key[r] ^ prefix) & high) == 0u) && (key[r] & bm)) ++loc;
        if (loc) atomicAdd(&cnt, loc);
        __syncthreads();
        int c = cnt;
        if (c >= need) prefix |= bm; else need -= c;
        __syncthreads();
    }
    if (tid == 0) wpos = 0;
    __syncthreads();
    int* dst = spages + (size_t)h * TOPK;
#pragma unroll
    for (int r = 0; r < 8; ++r)
        if (key[r] > prefix) { int p = atomicAdd(&wpos, 1); dst[p] = col[r]; }
    __syncthreads();
#pragma unroll
    for (int r = 0; r < 8; ++r)
        if (key[r] == prefix) { int p = atomicAdd(&wpos, 1); if (p < TOPK) dst[p] = col[r]; }
}

// =====================================================================
// Gather attention with flash-split softmax.
// Grid (32 heads x 8 chunks), 128 threads = 4 waves; each wave owns 8
// pages. Per page: stage 16x128 K tile to padded LDS (coalesced 512B
// rows), then a chain of 32 V_WMMA_F32_16X16X4_F32 computes the 16
// logits (B = q chunk replicated across all 16 columns; column N=0 is
// extracted from lane 0 / lane 16). V accumulated scalar (coalesced).
// =====================================================================
constexpr int KT_PAD = Dh + 4;   // 132 floats: row stride 528B, bank-conflict free

__global__ __launch_bounds__(128)
void attn_kernel(const float* __restrict__ kc, const float* __restrict__ vc,
                 const float* __restrict__ ws, const int* __restrict__ spages,
                 float* __restrict__ part)
{
    const int head  = blockIdx.x;
    const int chunk = blockIdx.y;
    const int tid   = threadIdx.x;
    const int wave  = tid >> 5;
    const int lane  = tid & 31;
    const int kvh   = head >> 2;      // kvmap = h // 4

    __shared__ float Kt[4][PAGE * KT_PAD];
    __shared__ float qs[Dh];
    __shared__ float mrg_m[4], mrg_l[4];
    __shared__ float mrg_acc[4][Dh];

    qs[tid] = ws[OFF_Q + (size_t)head * Dh + tid];
    __syncthreads();

    const float* knew = ws + OFF_K + (size_t)kvh * Dh;
    const float* vnew = ws + OFF_V + (size_t)kvh * Dh;

    float m = -INFINITY, l = 0.f;
    float4 acc = make_float4(0.f, 0.f, 0.f, 0.f);
    const int krow = (lane & 16) ? 2 : 0;        // K-rows held by this lane half
    const int arow = (lane & 15) * KT_PAD;       // A fragment row (M = token)
    float* kt = Kt[wave];

    for (int pi = 0; pi < 8; ++pi) {
        int page = spages[(size_t)head * TOPK + chunk * 32 + wave * 8 + pi];
        // --- stage K tile: one full 512B token row per iteration ---
#pragma unroll 4
        for (int t = 0; t < PAGE; ++t) {
            int tok = page * PAGE + t;
            const float* src = (tok == NEWTOK) ? knew
                               : kc + (size_t)tok * KVDIM + kvh * Dh;
            float4 v = *(const float4*)(src + lane * 4);
            *(float4*)(kt + t * KT_PAD + lane * 4) = v;
        }
        asm volatile("s_wait_dscnt 0" ::: "memory");  // tile visible before frag reads
        // --- 32-step f32 WMMA chain over D=128 ---
        vf8 cfrag = {};
#pragma unroll
        for (int ch = 0; ch < Dh / 4; ++ch) {
            vf2 a = *(const vf2*)(kt + arow + ch * 4 + krow);
            vf2 b = *(const vf2*)(qs + ch * 4 + krow);
            cfrag = __builtin_amdgcn_wmma_f32_16x16x4_f32(
                false, a, false, b, (short)0, cfrag, false, false);
        }
        // --- extract logits: C[M][0] lives in lane 0 (M=0..7) / lane 16 (M=8..15) ---
        float att[PAGE];
#pragma unroll
        for (int i = 0; i < 8; ++i) {
            att[i]     = __shfl(cfrag[i], 0, 32);
            att[i + 8] = __shfl(cfrag[i], 16, 32);
        }
        // --- online softmax + V accumulation ---
        float pm = m;
#pragma unroll
        for (int t = 0; t < PAGE; ++t) pm = fmaxf(pm, att[t] * SM_SCALE);
        float corr = __expf(m - pm);
        l *= corr; acc.x *= corr; acc.y *= corr; acc.z *= corr; acc.w *= corr;
        m = pm;
#pragma unroll 4
        for (int t = 0; t < PAGE; ++t) {
            float w = __expf(att[t] * SM_SCALE - m);
            l += w;
            int tok = page * PAGE + t;
            const float* vsrc = (tok == NEWTOK) ? vnew
                                : vc + (size_t)tok * KVDIM + kvh * Dh;
            float4 vv = *(const float4*)(vsrc + lane * 4);
            acc.x = fmaf(w, vv.x, acc.x);
            acc.y = fmaf(w, vv.y, acc.y);
            acc.z = fmaf(w, vv.z, acc.z);
            acc.w = fmaf(w, vv.w, acc.w);
        }
    }
    // --- merge the 4 waves of this block ---
    if (lane == 0) { mrg_m[wave] = m; mrg_l[wave] = l; }
    *(float4*)&mrg_acc[wave][lane * 4] = acc;
    __syncthreads();
    float M = fmaxf(fmaxf(mrg_m[0], mrg_m[1]), fmaxf(mrg_m[2], mrg_m[3]));
    float L = 0.f, val = 0.f;
#pragma unroll
    for (int w = 0; w < 4; ++w) {
        float e = __expf(mrg_m[w] - M);
        L   = fmaf(mrg_l[w], e, L);
        val = fmaf(mrg_acc[w][tid], e, val);
    }
    float* pb = part + (size_t)(head * 8 + chunk) * (Dh + 2);
    if (tid == 0) { pb[0] = M; pb[1] = L; }
    pb[2 + tid] = val;
}

// =====================================================================
// Merge the 8 chunk-partials per head; emit normalized attention output.
// =====================================================================
__global__ __launch_bounds__(128)
void merge_heads(const float* __restrict__ part, float* __restrict__ attn_out)
{
    const int h = blockIdx.x;
    const int d = threadIdx.x;   // 128
    float M = -INFINITY;
#pragma unroll
    for (int c = 0; c < 8; ++c)
        M = fmaxf(M, part[(size_t)(h * 8 + c) * (Dh + 2)]);
    float L = 0.f, val = 0.f;
#pragma unroll
    for (int c = 0; c < 8; ++c) {
        const float* pb = part + (size_t)(h * 8 + c) * (Dh + 2);
        float e = __expf(pb[0] - M);
        L   = fmaf(pb[1], e, L);
        val = fmaf(pb[2 + d], e, val);
    }
    attn_out[(size_t)h * Dh + d] = val / L;
}

// =====================================================================
extern "C" void kernel_launch(void* const* d_in, const int* in_sizes, int n_in,
                              void* d_out, int out_size, void* d_ws, size_t ws_size,
                              hipStream_t stream)
{
    (void)in_sizes; (void)n_in; (void)out_size; (void)ws_size;
    const float* x  = (const float*)d_in[0];
    const float* kc = (const float*)d_in[1];
    const float* vc = (const float*)d_in[2];
    const float* Wq = (const float*)d_in[3];
    const float* Wk = (const float*)d_in[4];
    const float* Wv = (const float*)d_in[5];
    const float* Wo = (const float*)d_in[6];
    float* out = (float*)d_out;
    float* ws  = (float*)d_ws;

    float* gpart    = ws + OFF_GPART;
    float* opart    = ws + OFF_OPART;
    float* scores   = ws + OFF_SCORES;
    int*   spages   = (int*)(ws + OFF_SPAGES);
    float* part     = ws + OFF_PART;
    float* attn_out = ws + OFF_ATTN;

    // q/k/v projections (split-K partials -> deterministic reduce)
    gemv_partial<<<dim3(HID / 256, GEMV_SPLITS), 256, 0, stream>>>(x, Wq, gpart, HID, QKV_STRIDE, 0);
    gemv_partial<<<dim3(KVDIM / 256, GEMV_SPLITS), 256, 0, stream>>>(x, Wk, gpart, KVDIM, QKV_STRIDE, HID);
    gemv_partial<<<dim3(KVDIM / 256, GEMV_SPLITS), 256, 0, stream>>>(x, Wv, gpart, KVDIM, QKV_STRIDE, HID + KVDIM);
    reduce_splits<<<QKV_STRIDE / 256, 256, 0, stream>>>(gpart, ws + OFF_Q, QKV_STRIDE, QKV_STRIDE);

    rope_kernel<<<((H + KV) * (Dh / 2) + 255) / 256, 256, 0, stream>>>(ws);
    page_score_kernel<<<NPAGES, 256, 0, stream>>>(kc, ws, scores);
    topk_kernel<<<H, 256, 0, stream>>>(scores, spages);
    attn_kernel<<<dim3(H, 8), 128, 0, stream>>>(kc, vc, ws, spages, part);
    merge_heads<<<H, Dh, 0, stream>>>(part, attn_out);

    // output projection
    gemv_partial<<<dim3(HID / 256, GEMV_SPLITS), 256, 0, stream>>>(attn_out, Wo, opart, HID, HID, 0);
    reduce_splits<<<HID / 256, 256, 0, stream>>>(opart, out, HID, HID);
}